// GraphAttentionLayer_16999480558111
// MI455X (gfx1250) — compile-verified
//
#include <hip/hip_runtime.h>
#include <hip/hip_bf16.h>
#include <math.h>

#define ALPHA 0.2f

typedef __attribute__((ext_vector_type(2))) float v2f;
typedef __attribute__((ext_vector_type(8))) float v8f;

// ---------------------------------------------------------------------------
// Kernel 1: h = X @ W  via V_WMMA_F32_16X16X4_F32.
// X: (2048,128) row-major, W: (128,64) row-major, H: (2048,64) row-major.
// One wave (32 lanes) per 16x16 output tile; K swept 128/4 = 32 WMMA ops.
//
// VGPR layouts (ISA 7.12.2, wave32):
//   A (16x4 f32):  lane<16 -> M=lane,    v0=K0, v1=K1
//                  lane>=16 -> M=lane-16, v0=K2, v1=K3
//   B (4x16 f32):  lane<16 -> N=lane,    v0=K0, v1=K1
//                  lane>=16 -> N=lane-16, v0=K2, v1=K3
//   C/D (16x16):   vgpr r: lane<16 -> M=r, N=lane ; lane>=16 -> M=r+8
// ---------------------------------------------------------------------------
__global__ __launch_bounds__(32) void gat_h_gemm_wmma(
    const float* __restrict__ X,   // (2048,128)
    const float* __restrict__ W,   // (128,64)
    float* __restrict__ H)         // (2048,64)
{
  const int tileM = blockIdx.x;          // 0..127
  const int tileN = blockIdx.y;          // 0..3
  const int lane  = threadIdx.x;         // 0..31
  const int half  = lane >> 4;           // 0 | 1
  const int l15   = lane & 15;

  const int rowA = tileM * 16 + l15;     // A-matrix M index for this lane
  const int colD = tileN * 16 + l15;     // B/C/D N index for this lane

  const float* __restrict__ xrow = X + rowA * 128;

  v8f acc = {};
#pragma unroll
  for (int k0 = 0; k0 < 128; k0 += 4) {
    const int k = k0 + 2 * half;
    v2f a, b;
    a.x = xrow[k];
    a.y = xrow[k + 1];
    b.x = W[k * 64 + colD];
    b.y = W[(k + 1) * 64 + colD];
    // (neg_a, A, neg_b, B, c_mod, C, reuse_a, reuse_b)
    acc = __builtin_amdgcn_wmma_f32_16x16x4_f32(
        false, a, false, b, (short)0, acc, false, false);
  }

#pragma unroll
  for (int r = 0; r < 8; ++r) {
    const int row = tileM * 16 + r + 8 * half;
    H[row * 64 + colD] = acc[r];
  }
}

// ---------------------------------------------------------------------------
// Kernel 2: collapsed attention + ELU. One 256-thread block per batch.
//
//   s[r]  = h[r] . (a_lo + a_hi)                 (drives rows i<256)
//   u[r]  = h[r] . a_lo ; v[r] = h[r] . a_hi
//   t[j]  = u[(2j)%512] + v[(2j)%512 + 1]        (drives rows i>=256)
//   Slo/Shi[c] = column sums of h rows [0,256)/[256,512)
//   rows i<256 : two-value softmax of L(s[2i]), L(s[2i+1]) -> w0,w1
//                out = elu(w0*Slo + w1*Shi)
//   rows i>=256: alpha = softmax_j(L(t[j])); P = sum_j alpha_j h[j]
//                out = elu(P)  (broadcast to all 256 rows)
// ---------------------------------------------------------------------------
__global__ __launch_bounds__(256) void gat_post(
    const float* __restrict__ H,   // (4,512,64)
    const float* __restrict__ A,   // (128,)
    float* __restrict__ OUT)       // (4,512,64)
{
  const int b   = blockIdx.x;
  const int tid = threadIdx.x;     // 0..255
  const float* __restrict__ h = H + b * 512 * 64;
  float* __restrict__ o       = OUT + b * 512 * 64;

  __shared__ float s_a[128];
  __shared__ float s_s[512];
  __shared__ float s_u[512];
  __shared__ float s_v[513];       // +1 pad: reads r0+1, r0<=510
  __shared__ float s_alpha[512];
  __shared__ float s_Slo[64];
  __shared__ float s_Shi[64];
  __shared__ float s_P[64];
  __shared__ float s_red[256];

  if (tid < 128) s_a[tid] = A[tid];
  __syncthreads();

  // Per-row dot products against a_lo / a_hi.
  for (int r = tid; r < 512; r += 256) {
    const float* hr = h + r * 64;
    float du = 0.f, dv = 0.f;
#pragma unroll 8
    for (int c = 0; c < 64; ++c) {
      const float x = hr[c];
      du += x * s_a[c];
      dv += x * s_a[c + 64];
    }
    s_u[r] = du;
    s_v[r] = dv;
    s_s[r] = du + dv;
  }

  // Column half-sums (reads only global h; no hazard with the loop above).
  if (tid < 128) {
    const int c    = tid & 63;
    const int base = (tid < 64) ? 0 : 256;
    float acc = 0.f;
    for (int r = 0; r < 256; ++r) acc += h[(base + r) * 64 + c];
    if (tid < 64) s_Slo[c] = acc; else s_Shi[c] = acc;
  }
  __syncthreads();

  // t[j] and its leaky-relu; 512-way softmax (2 elements per thread).
  float lt[2];
#pragma unroll
  for (int q = 0; q < 2; ++q) {
    const int j  = tid + q * 256;
    const int r0 = (2 * j) & 511;
    const float t = s_u[r0] + s_v[r0 + 1];
    lt[q] = (t > 0.f) ? t : ALPHA * t;
  }
  s_red[tid] = fmaxf(lt[0], lt[1]);
  __syncthreads();
  for (int off = 128; off > 0; off >>= 1) {
    if (tid < off) s_red[tid] = fmaxf(s_red[tid], s_red[tid + off]);
    __syncthreads();
  }
  const float m = s_red[0];
  __syncthreads();
  const float e0 = __expf(lt[0] - m);
  const float e1 = __expf(lt[1] - m);
  s_red[tid] = e0 + e1;
  __syncthreads();
  for (int off = 128; off > 0; off >>= 1) {
    if (tid < off) s_red[tid] += s_red[tid + off];
    __syncthreads();
  }
  const float Z = s_red[0];
  __syncthreads();
  s_alpha[tid]       = e0 / Z;
  s_alpha[tid + 256] = e1 / Z;
  __syncthreads();

  // P[c] = sum_j alpha[j] * h[j,c]
  if (tid < 64) {
    float acc = 0.f;
    for (int j = 0; j < 512; ++j) acc += s_alpha[j] * h[j * 64 + tid];
    s_P[tid] = (acc > 0.f) ? acc : (__expf(acc) - 1.f);   // fused ELU
  }
  __syncthreads();

  // Rows i < 256: per-row two-value softmax, blend of the half-sums, ELU.
  {
    const int i = tid;
    const float sv0 = s_s[2 * i];
    const float sv1 = s_s[2 * i + 1];
    const float v0  = (sv0 > 0.f) ? sv0 : ALPHA * sv0;
    const float v1  = (sv1 > 0.f) ? sv1 : ALPHA * sv1;
    const float mm  = fmaxf(v0, v1);
    const float ee0 = __expf(v0 - mm);
    const float ee1 = __expf(v1 - mm);
    const float den = 256.f * (ee0 + ee1);
    const float w0  = ee0 / den;
    const float w1  = ee1 / den;
#pragma unroll 8
    for (int c = 0; c < 64; ++c) {
      const float hp = w0 * s_Slo[c] + w1 * s_Shi[c];
      o[i * 64 + c] = (hp > 0.f) ? hp : (__expf(hp) - 1.f);
    }
  }

  // Rows i >= 256: broadcast elu(P).
  for (int idx = tid; idx < 256 * 64; idx += 256) {
    const int i = 256 + (idx >> 6);
    const int c = idx & 63;
    o[i * 64 + c] = s_P[c];
  }
}

// ---------------------------------------------------------------------------
extern "C" void kernel_launch(void* const* d_in, const int* in_sizes, int n_in,
                              void* d_out, int out_size, void* d_ws, size_t ws_size,
                              hipStream_t stream) {
  const float* X = (const float*)d_in[0];   // input_feature (4,512,128)
  // d_in[1] = adj (int32, dead code in the reference)
  const float* W = (const float*)d_in[2];   // (128,64)
  const float* A = (const float*)d_in[3];   // (128,1)
  float* OUT = (float*)d_out;               // (4,512,64)
  float* H   = (float*)d_ws;                // (2048,64) scratch = 512 KB

  dim3 gemmGrid(128, 4);                    // 16x16 tiles of (2048 x 64)
  gat_h_gemm_wmma<<<gemmGrid, 32, 0, stream>>>(X, W, H);

  gat_post<<<4, 256, 0, stream>>>(H, A, OUT);
}